// GCN_34832184771167
// MI455X (gfx1250) — compile-verified
//
#include <hip/hip_runtime.h>
#include <hip/hip_bf16.h>

typedef __attribute__((ext_vector_type(2))) float v2f;
typedef __attribute__((ext_vector_type(8))) float v8f;

#define DIN 128
#define DOUT 128
#define ROWS_PER_BLOCK 32
// Padded LDS row strides (in floats) chosen so the two half-wave K-offsets of
// the f32 WMMA fragment reads land on disjoint bank ranges (64 banks):
#define FS 132   // feature tile stride: lane m reads m*FS + k -> banks 4m+k, conflict-free
#define WS 144   // weight stride: rows k and k+2 differ by 288 mod 64 = 32 banks -> disjoint halves
#define LDS_W_FLOATS (DIN * WS)
#define LDS_F_FLOATS (ROWS_PER_BLOCK * FS)
#define LDS_BYTES ((LDS_W_FLOATS + LDS_F_FLOATS) * 4)

// ---------------------------------------------------------------------------
// Kernel 1: h = features @ weight (exact fp32 via V_WMMA_F32_16X16X4_F32),
// plus epilogue seed: out = h * skip + bias. h kept in d_ws (L2-resident).
// Block = 256 threads = 8 waves. Block covers 32 rows; wave w owns cols
// [16w, 16w+16) and both 16-row M-tiles.
// ---------------------------------------------------------------------------
__global__ void gcn_gemm_kernel(const float* __restrict__ features,
                                const float* __restrict__ weight,
                                const float* __restrict__ bias,
                                const float* __restrict__ skip,
                                float* __restrict__ h_out,
                                float* __restrict__ out_init,
                                int n_nodes) {
  extern __shared__ float smem[];
  float* lds_w = smem;                  // [128][WS]
  float* lds_f = smem + LDS_W_FLOATS;   // [32][FS]

  const int t  = threadIdx.x;
  const int rb = blockIdx.x * ROWS_PER_BLOCK;

  // Cooperative load: full 128x128 weight (4096 float4) + 32x128 feature tile
  // (1024 float4) into padded LDS. float4 stores stay 16B aligned (FS*4=528,
  // WS*4=576 are multiples of 16).
  {
    const float4* w4 = (const float4*)weight;
#pragma unroll
    for (int s = 0; s < 16; ++s) {
      int idx = t + s * 256;            // 0..4095
      int row = idx >> 5;               // /32 float4 per row
      int c4  = idx & 31;
      float4 v = w4[idx];
      *(float4*)&lds_w[row * WS + c4 * 4] = v;
    }
#pragma unroll
    for (int s = 0; s < 4; ++s) {
      int idx = t + s * 256;            // 0..1023
      int row = idx >> 5;
      int c4  = idx & 31;
      int grow = rb + row;
      if (grow > n_nodes - 1) grow = n_nodes - 1;   // clamp (no-op here)
      float4 v = *(const float4*)(features + (size_t)grow * DIN + c4 * 4);
      *(float4*)&lds_f[row * FS + c4 * 4] = v;
    }
  }
  __syncthreads();

  const int wave = t >> 5;        // 0..7 -> N tile
  const int lane = t & 31;
  const int half = lane >> 4;     // 0/1 : which K-pair this lane holds
  const int l16  = lane & 15;
  const int koff = half * 2;
  const int nb   = wave * 16;

  v8f acc0 = {};   // rows rb+0..15
  v8f acc1 = {};   // rows rb+16..31

#pragma unroll 8
  for (int kb = 0; kb < DIN; kb += 4) {
    const int k = kb + koff;
    // A fragments (16x4 f32): lane holds row l16, K = {k, k+1} -> ds_load_b64
    v2f a0 = *(const v2f*)&lds_f[l16 * FS + k];
    v2f a1 = *(const v2f*)&lds_f[(16 + l16) * FS + k];
    // B fragment (4x16 f32): lane holds col nb+l16, K = {k, k+1}
    v2f b;
    b.x = lds_w[(k + 0) * WS + nb + l16];
    b.y = lds_w[(k + 1) * WS + nb + l16];
    acc0 = __builtin_amdgcn_wmma_f32_16x16x4_f32(false, a0, false, b,
                                                 (short)0, acc0, false, false);
    acc1 = __builtin_amdgcn_wmma_f32_16x16x4_f32(false, a1, false, b,
                                                 (short)0, acc1, false, false);
  }

  // D layout: VGPR r -> row half*8 + r, col = nb + l16
  const int col = nb + l16;
  const float sk = skip[col];
  const float bs = bias[col];

#pragma unroll
  for (int r = 0; r < 8; ++r) {
    int m0 = rb + half * 8 + r;
    int m1 = m0 + 16;
    float hv0 = acc0[r];
    float hv1 = acc1[r];
    if (m0 < n_nodes) {
      h_out[(size_t)m0 * DOUT + col]    = hv0;
      out_init[(size_t)m0 * DOUT + col] = hv0 * sk + bs;
    }
    if (m1 < n_nodes) {
      h_out[(size_t)m1 * DOUT + col]    = hv1;
      out_init[(size_t)m1 * DOUT + col] = hv1 * sk + bs;
    }
  }
}

// ---------------------------------------------------------------------------
// Kernel 2: COO SpMM scatter. One wave per edge; lane i handles the float4 at
// feature offset 4i. h rows (51.2 MB) and out (51.2 MB) both fit in the 192 MB
// L2, so gathers + fp32 atomics are L2-resident.
// ---------------------------------------------------------------------------
__global__ void gcn_edge_kernel(const int* __restrict__ erow,
                                const int* __restrict__ ecol,
                                const float* __restrict__ evalp,
                                const float* __restrict__ h,
                                float* __restrict__ out,
                                int n_edges) {
  int gid  = blockIdx.x * blockDim.x + threadIdx.x;
  int e    = gid >> 5;
  int lane = threadIdx.x & 31;
  if (e >= n_edges) return;
  int   r = erow[e];
  int   c = ecol[e];
  float v = evalp[e];
  float4 g = *(const float4*)(h + (size_t)c * DOUT + lane * 4);
  float* dst = out + (size_t)r * DOUT + lane * 4;
  atomicAdd(dst + 0, v * g.x);
  atomicAdd(dst + 1, v * g.y);
  atomicAdd(dst + 2, v * g.z);
  atomicAdd(dst + 3, v * g.w);
}

// ---------------------------------------------------------------------------
// Kernel 3: in-place SELU over d_out.
// ---------------------------------------------------------------------------
__global__ void gcn_selu_kernel(float* __restrict__ out, int n4) {
  int i = blockIdx.x * blockDim.x + threadIdx.x;
  if (i >= n4) return;
  const float alpha = 1.6732632423543772f;
  const float scale = 1.0507009873554805f;
  float4* p = (float4*)out;
  float4 x = p[i];
  x.x = x.x > 0.0f ? scale * x.x : scale * alpha * expm1f(x.x);
  x.y = x.y > 0.0f ? scale * x.y : scale * alpha * expm1f(x.y);
  x.z = x.z > 0.0f ? scale * x.z : scale * alpha * expm1f(x.z);
  x.w = x.w > 0.0f ? scale * x.w : scale * alpha * expm1f(x.w);
  p[i] = x;
}

extern "C" void kernel_launch(void* const* d_in, const int* in_sizes, int n_in,
                              void* d_out, int out_size, void* d_ws, size_t ws_size,
                              hipStream_t stream) {
  const float* features = (const float*)d_in[0];
  const int*   erow     = (const int*)d_in[1];
  const int*   ecol     = (const int*)d_in[2];
  const float* evalp    = (const float*)d_in[3];
  const float* weight   = (const float*)d_in[4];
  const float* bias     = (const float*)d_in[5];
  const float* skip     = (const float*)d_in[6];
  float* out = (float*)d_out;
  float* h   = (float*)d_ws;            // n_nodes*128 floats (51.2 MB)

  const int n_nodes = in_sizes[0] / DIN;   // 100000
  const int n_edges = in_sizes[1];         // 1600000

  // K1: projection + out seed
  int gemm_blocks = (n_nodes + ROWS_PER_BLOCK - 1) / ROWS_PER_BLOCK;
  gcn_gemm_kernel<<<gemm_blocks, 256, LDS_BYTES, stream>>>(
      features, weight, bias, skip, h, out, n_nodes);

  // K2: edge scatter (one wave32 per edge)
  long long edge_threads = (long long)n_edges * 32;
  int edge_blocks = (int)((edge_threads + 255) / 256);
  gcn_edge_kernel<<<edge_blocks, 256, 0, stream>>>(erow, ecol, evalp, h, out, n_edges);

  // K3: SELU epilogue
  int n4 = n_nodes * DOUT / 4;
  gcn_selu_kernel<<<(n4 + 255) / 256, 256, 0, stream>>>(out, n4);
}